// TemporalGAT_44229573214311
// MI455X (gfx1250) — compile-verified
//
#include <hip/hip_runtime.h>
#include <math.h>

// ---------------------------------------------------------------------------
// TemporalGAT on MI455X (gfx1250): 2-layer GAT + dense skips.
// Dense GEMMs via V_WMMA_F32_16X16X4_F32 (fp32 exact path, templated shapes,
// 16x32 C strip per wave); edge softmax + message scatter via coalesced
// global float atomics (the HBM-bound part).
// ---------------------------------------------------------------------------

typedef __attribute__((ext_vector_type(2))) float v2f;
typedef __attribute__((ext_vector_type(8))) float v8f;

#define NEG_SLOPE 0.2f

// ---------------- WMMA fp32 GEMM: C[M x Nc] = A[M x K] @ B[K x Nc] ----------
// Row-major. One 16x32 C strip per wave32 (two accumulators sharing the A
// fragment); K stepped by 4 per WMMA pair. Compile-time K/Nc so B-row offsets
// fold into immediate load offsets.
// ISA layouts (CDNA5 ISA):
//   A 16x4 : lanes0-15 row M, {K0,K1} in v[0],v[1]; lanes16-31 same rows {K2,K3}
//   B 4x16 : VGPR0 = rows {K0 | K2}, VGPR1 = rows {K1 | K3} across lane halves
//   C 16x16: VGPR v = row v (lanes0-15) / row v+8 (lanes16-31)
template <int K, int Nc>
__global__ __launch_bounds__(256) void gemm_wmma_f32(
    const float* __restrict__ A, const float* __restrict__ B,
    float* __restrict__ C, int M) {
  constexpr int PAIRS = Nc / 32;        // 16x32 strips per tile-row
  const int lane = threadIdx.x & 31;
  const int wave = blockIdx.x * (blockDim.x >> 5) + (threadIdx.x >> 5);
  const int tM = wave / PAIRS;
  const int tp = wave - tM * PAIRS;
  const int l15 = lane & 15;
  const int half = lane >> 4;           // 0: lanes 0-15, 1: lanes 16-31
  int arow = tM * 16 + l15;
  if (arow >= M) arow = M - 1;          // wave-uniform tiling keeps EXEC full
  const int bcol = tp * 32 + l15;
  const int koff = half * 2;
  const float* Ap = A + (size_t)arow * K + koff;
  const float* Bp = B + (size_t)koff * Nc + bcol;
  v8f acc0 = {}, acc1 = {};
#pragma unroll 8
  for (int k = 0; k < K; k += 4) {
    v2f a = *(const v2f*)(Ap + k);      // A[row][k+koff], A[row][k+koff+1]
    v2f b0, b1;
    b0.x = Bp[k * Nc];                  // B[k+koff  ][col]
    b0.y = Bp[(k + 1) * Nc];            // B[k+koff+1][col]
    b1.x = Bp[k * Nc + 16];             // B[k+koff  ][col+16]
    b1.y = Bp[(k + 1) * Nc + 16];       // B[k+koff+1][col+16]
    acc0 = __builtin_amdgcn_wmma_f32_16x16x4_f32(
        false, a, false, b0, (short)0, acc0, false, false);
    acc1 = __builtin_amdgcn_wmma_f32_16x16x4_f32(
        false, a, false, b1, (short)0, acc1, false, false);
  }
  const int crow0 = tM * 16 + half * 8;
#pragma unroll
  for (int v = 0; v < 8; ++v) {
    int r = crow0 + v;
    if (r < M) {
      C[(size_t)r * Nc + bcol] = acc0[v];
      C[(size_t)r * Nc + bcol + 16] = acc1[v];
    }
  }
}

// ------------- a_src/a_dst: wave-per-(node,head) dot-product reduction ------
__global__ __launch_bounds__(256) void attn_dot(
    const float* __restrict__ h, const float* __restrict__ att_src,
    const float* __restrict__ att_dst, float* __restrict__ a_src,
    float* __restrict__ a_dst, int n_pairs, int heads, int ch) {
  const int wave = blockIdx.x * (blockDim.x >> 5) + (threadIdx.x >> 5);
  const int lane = threadIdx.x & 31;
  if (wave >= n_pairs) return;
  const int node = wave / heads;
  const int head = wave - node * heads;
  const float* hp = h + (size_t)node * heads * ch + head * ch;
  const float* as = att_src + head * ch;
  const float* ad = att_dst + head * ch;
  float ss = 0.f, sd = 0.f;
  for (int c = lane; c < ch; c += 32) {
    float v = hp[c];
    ss += v * as[c];
    sd += v * ad[c];
  }
#pragma unroll
  for (int off = 16; off > 0; off >>= 1) {
    ss += __shfl_down(ss, off, 32);
    sd += __shfl_down(sd, off, 32);
  }
  if (lane == 0) { a_src[wave] = ss; a_dst[wave] = sd; }
}

// ----------------------------- fill helper ---------------------------------
__global__ void fill_f32(float* __restrict__ p, float v, long long n) {
  long long i = (long long)blockIdx.x * blockDim.x + threadIdx.x;
  if (i < n) p[i] = v;
}

// Sign-aware float atomic max via integer punning (works for mixed signs).
__device__ __forceinline__ void atomicMaxFloat(float* addr, float val) {
  if (val >= 0.f) atomicMax((int*)addr, __float_as_int(val));
  else            atomicMin((unsigned int*)addr, __float_as_uint(val));
}

// -------- edge logits + segment max (self-loops appended at e >= E) ---------
__global__ void edge_logits(const int* __restrict__ src, const int* __restrict__ dst,
                            const float* __restrict__ a_src, const float* __restrict__ a_dst,
                            float* __restrict__ elog, float* __restrict__ m,
                            int E, int Ntot, int hlog2) {
  const int H = 1 << hlog2;
  int idx = blockIdx.x * blockDim.x + threadIdx.x;
  int tot = (E + Ntot) << hlog2;
  if (idx >= tot) return;
  int e = idx >> hlog2, hh = idx & (H - 1);
  int sN, dN;
  if (e < E) { sN = src[e]; dN = dst[e]; } else { sN = dN = e - E; }
  float l = a_src[(sN << hlog2) + hh] + a_dst[(dN << hlog2) + hh];
  l = l > 0.f ? l : NEG_SLOPE * l;
  elog[idx] = l;
  atomicMaxFloat(&m[(dN << hlog2) + hh], l);
}

// ------------------------- exp + segment sum --------------------------------
__global__ void edge_exp_sum(const int* __restrict__ dst, const float* __restrict__ m,
                             float* __restrict__ elog, float* __restrict__ ssum,
                             int E, int Ntot, int hlog2) {
  const int H = 1 << hlog2;
  int idx = blockIdx.x * blockDim.x + threadIdx.x;
  int tot = (E + Ntot) << hlog2;
  if (idx >= tot) return;
  int e = idx >> hlog2, hh = idx & (H - 1);
  int dN = (e < E) ? dst[e] : (e - E);
  float ex = expf(elog[idx] - m[(dN << hlog2) + hh]);
  elog[idx] = ex;
  atomicAdd(&ssum[(dN << hlog2) + hh], ex);
}

// ------------------- normalize: elog <- alpha in place ----------------------
__global__ void edge_norm(const int* __restrict__ dst, const float* __restrict__ ssum,
                          float* __restrict__ elog, int E, int Ntot, int hlog2) {
  const int H = 1 << hlog2;
  int idx = blockIdx.x * blockDim.x + threadIdx.x;
  int tot = (E + Ntot) << hlog2;
  if (idx >= tot) return;
  int e = idx >> hlog2, hh = idx & (H - 1);
  int dN = (e < E) ? dst[e] : (e - E);
  elog[idx] = elog[idx] / (ssum[(dN << hlog2) + hh] + 1e-16f);
}

// --------- message scatter: out[dst] += alpha * h[src], coalesced -----------
__global__ void scatter_msg(const int* __restrict__ src, const int* __restrict__ dst,
                            const float* __restrict__ alpha, const float* __restrict__ hfeat,
                            float* __restrict__ out, int E, int Ntot,
                            int hlog2, int chlog2) {
  const int flog2 = hlog2 + chlog2;
  const int F = 1 << flog2;
  long long idx = (long long)blockIdx.x * blockDim.x + threadIdx.x;
  long long tot = (long long)(E + Ntot) << flog2;
  if (idx >= tot) return;
  int e = (int)(idx >> flog2);
  int f = (int)idx & (F - 1);
  int sN, dN;
  if (e < E) { sN = src[e]; dN = dst[e]; } else { sN = dN = e - E; }
  float a = alpha[((long long)e << hlog2) + (f >> chlog2)];
  atomicAdd(&out[((size_t)dN << flog2) + f], a * hfeat[((size_t)sN << flog2) + f]);
}

// ------------- layer-1 combine: h = ELU(gat + b1 + lin + lin_b) -------------
__global__ void combine_elu(const float* __restrict__ gat, const float* __restrict__ b1,
                            const float* __restrict__ lin, const float* __restrict__ lin_b,
                            float* __restrict__ hout, long long tot, int fmask) {
  long long idx = (long long)blockIdx.x * blockDim.x + threadIdx.x;
  if (idx >= tot) return;
  int f = (int)idx & fmask;
  float v = gat[idx] + b1[f] + lin[idx] + lin_b[f];
  hout[idx] = v > 0.f ? v : expm1f(v);
}

// ------------- finalize: out += gat2 + b2 + lin2_b (out has h@lin2_w) -------
__global__ void finalize(const float* __restrict__ gat2, const float* __restrict__ b2,
                         const float* __restrict__ lin2_b, float* __restrict__ out,
                         long long tot, int fmask) {
  long long idx = (long long)blockIdx.x * blockDim.x + threadIdx.x;
  if (idx >= tot) return;
  int f = (int)idx & fmask;
  out[idx] += gat2[idx] + b2[f] + lin2_b[f];
}

// ---------------------------------------------------------------------------
static inline int cdiv(long long a, int b) { return (int)((a + b - 1) / b); }

extern "C" void kernel_launch(void* const* d_in, const int* in_sizes, int n_in,
                              void* d_out, int out_size, void* d_ws, size_t ws_size,
                              hipStream_t stream) {
  const int IN = 128, HC = 256, OUT = 128, HEADS = 4, HID = 64;
  const int N = in_sizes[0] / IN;          // 50000
  const int E = in_sizes[1] / 2;           // 800000
  const int E2 = E + N;                    // with self-loops

  const float* x       = (const float*)d_in[0];
  const int*   ei      = (const int*)d_in[1];
  const int*   src     = ei;
  const int*   dstp    = ei + E;
  const float* W1      = (const float*)d_in[2];
  const float* att_s1  = (const float*)d_in[3];
  const float* att_d1  = (const float*)d_in[4];
  const float* b1      = (const float*)d_in[5];
  const float* lin1_w  = (const float*)d_in[6];
  const float* lin1_b  = (const float*)d_in[7];
  const float* W2      = (const float*)d_in[8];
  const float* att_s2  = (const float*)d_in[9];
  const float* att_d2  = (const float*)d_in[10];
  const float* b2      = (const float*)d_in[11];
  const float* lin2_w  = (const float*)d_in[12];
  const float* lin2_b  = (const float*)d_in[13];
  float* out = (float*)d_out;

  // -------- workspace layout (floats); total ~171 MB --------
  float* ws   = (float*)d_ws;
  float* h1   = ws;                         // N*256 : x@W1
  float* lin1 = h1 + (size_t)N * HC;        // N*256 : x@lin1_w, then h (post-ELU)
  float* scr  = lin1 + (size_t)N * HC;      // N*256 : L1 gat accum; L2: h2 | gat2
  float* asrc = scr + (size_t)N * HC;       // N*4
  float* adst = asrc + (size_t)N * HEADS;   // N*4
  float* mbuf = adst + (size_t)N * HEADS;   // N*4
  float* sbuf = mbuf + (size_t)N * HEADS;   // N*4
  float* elog = sbuf + (size_t)N * HEADS;   // E2*4
  (void)ws_size; (void)n_in; (void)out_size;

  // ============================ Layer 1 ============================
  // h1 = x @ W1 ; lin1 = x @ lin1_w   (M=N, K=128, Nc=256)
  {
    int pairs  = HC / 32;                             // 8 strips per tile-row
    int waves  = ((N + 15) / 16) * pairs;             // 25000
    int blocks = (waves + 7) / 8;
    gemm_wmma_f32<128, 256><<<blocks, 256, 0, stream>>>(x, W1, h1, N);
    gemm_wmma_f32<128, 256><<<blocks, 256, 0, stream>>>(x, lin1_w, lin1, N);
  }
  // attention coefficients per (node, head)
  {
    int pairs = N * HEADS;
    attn_dot<<<(pairs + 7) / 8, 256, 0, stream>>>(h1, att_s1, att_d1, asrc, adst,
                                                  pairs, HEADS, HID);
  }
  fill_f32<<<cdiv((long long)N * HEADS, 256), 256, 0, stream>>>(mbuf, -INFINITY, (long long)N * HEADS);
  fill_f32<<<cdiv((long long)N * HEADS, 256), 256, 0, stream>>>(sbuf, 0.f, (long long)N * HEADS);
  fill_f32<<<cdiv((long long)N * HC, 256), 256, 0, stream>>>(scr, 0.f, (long long)N * HC);
  {
    int tot = E2 * HEADS, hl = 2;
    edge_logits<<<cdiv(tot, 256), 256, 0, stream>>>(src, dstp, asrc, adst, elog, mbuf, E, N, hl);
    edge_exp_sum<<<cdiv(tot, 256), 256, 0, stream>>>(dstp, mbuf, elog, sbuf, E, N, hl);
    edge_norm<<<cdiv(tot, 256), 256, 0, stream>>>(dstp, sbuf, elog, E, N, hl);
    long long stot = (long long)E2 * HC;              // 217.6M
    scatter_msg<<<cdiv(stot, 256), 256, 0, stream>>>(src, dstp, elog, h1, scr, E, N, hl, 6);
  }
  // h = ELU(gat1 + b1 + lin1 + lin1_b)  -> stored into lin1 buffer
  combine_elu<<<cdiv((long long)N * HC, 256), 256, 0, stream>>>(
      scr, b1, lin1, lin1_b, lin1, (long long)N * HC, HC - 1);
  float* h = lin1;

  // ============================ Layer 2 ============================
  float* h2   = scr;                        // N*128
  float* gat2 = scr + (size_t)N * OUT;      // N*128
  {
    int pairs  = OUT / 32;                            // 4 strips per tile-row
    int waves  = ((N + 15) / 16) * pairs;             // 12500
    int blocks = (waves + 7) / 8;
    gemm_wmma_f32<256, 128><<<blocks, 256, 0, stream>>>(h, W2, h2, N);
    gemm_wmma_f32<256, 128><<<blocks, 256, 0, stream>>>(h, lin2_w, out, N);
  }
  attn_dot<<<(N + 7) / 8, 256, 0, stream>>>(h2, att_s2, att_d2, asrc, adst, N, 1, OUT);
  fill_f32<<<cdiv(N, 256), 256, 0, stream>>>(mbuf, -INFINITY, N);
  fill_f32<<<cdiv(N, 256), 256, 0, stream>>>(sbuf, 0.f, N);
  fill_f32<<<cdiv((long long)N * OUT, 256), 256, 0, stream>>>(gat2, 0.f, (long long)N * OUT);
  {
    int tot = E2, hl = 0;
    edge_logits<<<cdiv(tot, 256), 256, 0, stream>>>(src, dstp, asrc, adst, elog, mbuf, E, N, hl);
    edge_exp_sum<<<cdiv(tot, 256), 256, 0, stream>>>(dstp, mbuf, elog, sbuf, E, N, hl);
    edge_norm<<<cdiv(tot, 256), 256, 0, stream>>>(dstp, sbuf, elog, E, N, hl);
    long long stot = (long long)E2 * OUT;             // 108.8M
    scatter_msg<<<cdiv(stot, 256), 256, 0, stream>>>(src, dstp, elog, h2, gat2, E, N, hl, 7);
  }
  finalize<<<cdiv((long long)N * OUT, 256), 256, 0, stream>>>(
      gat2, b2, lin2_b, out, (long long)N * OUT, OUT - 1);
}